// Model_60979945668953
// MI455X (gfx1250) — compile-verified
//
#include <hip/hip_runtime.h>
#include <math.h>

typedef __attribute__((ext_vector_type(2))) float v2f;
typedef __attribute__((ext_vector_type(8))) float v8f;

#define N_NODES   50000
#define N_EDGES   800000
#define F_IN      256
#define HID       256
#define NHID      128
#define N_GRAPHS  128
#define MAX_K     512

__device__ __forceinline__ float selu_f(float x) {
    const float alpha = 1.6732632423543772f;
    const float scale = 1.0507009873554805f;
    return scale * (x > 0.f ? x : alpha * (expf(x) - 1.f));
}

__device__ __forceinline__ v8f wmma_f32(v2f a, v2f b, v8f c) {
    return __builtin_amdgcn_wmma_f32_16x16x4_f32(
        false, a, false, b, (short)0, c, false, false);
}

// Async global -> LDS copy (16B per lane), tracked by ASYNCcnt.
__device__ __forceinline__ void async_copy_b128(unsigned lds_off, const void* gaddr) {
    asm volatile("global_load_async_to_lds_b128 %0, %1, off"
                 :: "v"(lds_off), "v"(gaddr) : "memory");
}
__device__ __forceinline__ void wait_asynccnt0() {
    asm volatile("s_wait_asynccnt 0" ::: "memory");
}

// ---------------------------------------------------------------------------
// WMMA GEMM:  C[M,N] = act( rowscale[m] * (A[M,K] @ B[K,N]) + bias[n] )
// One block = one 16-row strip of A (async-staged to LDS) x full N.
// One wave  = 16x64 output (4 accumulators share each A fragment).
// Launch: grid = M/16, block = (N/64)*32.  N%64==0, K%4==0, K<=MAX_K.
// act: 0=none, 1=relu, 2=selu.  rowscale/bias may be null.
// ---------------------------------------------------------------------------
__global__ void wmma_gemm_f32(const float* __restrict__ A, int lda,
                              const float* __restrict__ B, int ldb,
                              float* __restrict__ C, int ldc,
                              const float* __restrict__ bias,
                              const float* __restrict__ rowscale,
                              int N, int K, int act)
{
    __shared__ float As[16 * MAX_K];               // 32 KB
    const int lane = threadIdx.x & 31;
    const int wave = threadIdx.x >> 5;
    const int row0 = blockIdx.x << 4;

    // ---- stage A strip [16 x K] into LDS via async b128 copies ----
    {
        const unsigned lds_base = (unsigned)(size_t)(void*)As;
        const int chunks = (16 * K) >> 2;          // float4 chunks
        const int kq = K >> 2;                     // chunks per row
        for (int cch = threadIdx.x; cch < chunks; cch += blockDim.x) {
            const int r  = cch / kq;
            const int kc = cch - r * kq;
            async_copy_b128(lds_base + (unsigned)(r * K + kc * 4) * 4u,
                            A + (size_t)(row0 + r) * lda + kc * 4);
        }
        wait_asynccnt0();
        __syncthreads();
    }

    const int lh  = lane >> 4;                     // lane half (0/1)
    const int l16 = lane & 15;
    const int colb = wave << 6;                    // this wave's 64-col base

    // A fragment: lane holds row M=l16; VGPR v holds K = k0 + 2*lh + v
    // B fragment: lane holds col N=l16; VGPR v holds K = k0 + 2*lh + v
    const float* __restrict__ Asl = &As[l16 * K];
    const float* __restrict__ Bp  = B + colb + l16;

    v8f c0 = {}, c1 = {}, c2 = {}, c3 = {};
    #pragma unroll 2
    for (int k0 = 0; k0 < K; k0 += 4) {
        const int kk = k0 + (lh << 1);
        v2f a;
        a.x = Asl[kk];
        a.y = Asl[kk + 1];
        const float* __restrict__ Br0 = Bp + (size_t)kk * ldb;
        const float* __restrict__ Br1 = Bp + (size_t)(kk + 1) * ldb;
        v2f b0, b1, b2, b3;
        b0.x = Br0[0];  b0.y = Br1[0];
        b1.x = Br0[16]; b1.y = Br1[16];
        b2.x = Br0[32]; b2.y = Br1[32];
        b3.x = Br0[48]; b3.y = Br1[48];
        c0 = wmma_f32(a, b0, c0);
        c1 = wmma_f32(a, b1, c1);
        c2 = wmma_f32(a, b2, c2);
        c3 = wmma_f32(a, b3, c3);
    }

    // C/D layout: VGPR v -> row = row0 + v + 8*lh, col = colb + 16*t + l16
    v8f acc[4] = {c0, c1, c2, c3};
    #pragma unroll
    for (int t = 0; t < 4; ++t) {
        const int col = colb + (t << 4) + l16;
        const float bv = bias ? bias[col] : 0.f;
        #pragma unroll
        for (int v = 0; v < 8; ++v) {
            const int row = row0 + v + (lh << 3);
            float val = acc[t][v];
            if (rowscale) val *= rowscale[row];
            val += bv;
            if (act == 1)      val = fmaxf(val, 0.f);
            else if (act == 2) val = selu_f(val);
            C[(size_t)row * ldc + col] = val;
        }
    }
}

// ---------------------------------------------------------------------------
// Degree / normalization
// ---------------------------------------------------------------------------
__global__ void deg_init(float* __restrict__ deg) {
    int i = blockIdx.x * blockDim.x + threadIdx.x;
    if (i < N_NODES) deg[i] = 1.0f;            // self-loop
}
__global__ void deg_count(const int* __restrict__ dst, float* __restrict__ deg) {
    int e = blockIdx.x * blockDim.x + threadIdx.x;
    if (e < N_EDGES) unsafeAtomicAdd(&deg[dst[e]], 1.0f);
}
__global__ void deg_rsqrt(float* __restrict__ deg) {
    int i = blockIdx.x * blockDim.x + threadIdx.x;
    if (i < N_NODES) deg[i] = rsqrtf(deg[i]);  // deg >= 1 always
}

// ---------------------------------------------------------------------------
// Aggregation:  acc[i] = g[i]*dinv[i]  (g already has dinv[src] folded in by
// the GEMM rowscale);  then  acc[dst] += g[src]*dinv[dst]  per edge.
// ---------------------------------------------------------------------------
__global__ void self_init(const float* __restrict__ g, const float* __restrict__ dinv,
                          float* __restrict__ acc) {
    int id = blockIdx.x * blockDim.x + threadIdx.x;        // N_NODES*64 quads
    if (id >= N_NODES * 64) return;
    const int i = id >> 6, q = id & 63;
    float4 v = ((const float4*)(g + (size_t)i * HID))[q];
    const float w = dinv[i];
    v.x *= w; v.y *= w; v.z *= w; v.w *= w;
    ((float4*)(acc + (size_t)i * HID))[q] = v;
}

__global__ void edge_scatter(const float* __restrict__ g, const float* __restrict__ dinv,
                             const int* __restrict__ src, const int* __restrict__ dst,
                             float* __restrict__ acc) {
    int id = blockIdx.x * blockDim.x + threadIdx.x;        // N_EDGES*64 quads
    if (id >= N_EDGES * 64) return;
    const int e = id >> 6, q = id & 63;
    const int s = src[e], d = dst[e];
    const float w = dinv[d];
    const float4 v = ((const float4*)(g + (size_t)s * HID))[q];
    float* p = acc + (size_t)d * HID + (q << 2);
    unsafeAtomicAdd(p + 0, v.x * w);
    unsafeAtomicAdd(p + 1, v.y * w);
    unsafeAtomicAdd(p + 2, v.z * w);
    unsafeAtomicAdd(p + 3, v.w * w);
}

__global__ void bias_selu(float* __restrict__ acc, const float* __restrict__ b) {
    int id = blockIdx.x * blockDim.x + threadIdx.x;        // N_NODES*64 quads
    if (id >= N_NODES * 64) return;
    const int i = id >> 6, q = id & 63;
    float4 v  = ((float4*)(acc + (size_t)i * HID))[q];
    const float4 bv = ((const float4*)b)[q];
    v.x = selu_f(v.x + bv.x);
    v.y = selu_f(v.y + bv.y);
    v.z = selu_f(v.z + bv.z);
    v.w = selu_f(v.w + bv.w);
    ((float4*)(acc + (size_t)i * HID))[q] = v;
}

// ---------------------------------------------------------------------------
// Global mean pool + head helpers
// ---------------------------------------------------------------------------
__global__ void zin_zero(float* __restrict__ zin, float* __restrict__ cnt) {
    int id = blockIdx.x * blockDim.x + threadIdx.x;
    if (id < N_GRAPHS * 4 * NHID) zin[id] = 0.f;
    if (id < N_GRAPHS) cnt[id] = 0.f;
}

__global__ void pool_scatter(const float* __restrict__ h, const int* __restrict__ batch,
                             float* __restrict__ zin, float* __restrict__ cnt) {
    int id = blockIdx.x * blockDim.x + threadIdx.x;        // N_NODES*64 quads
    if (id >= N_NODES * 64) return;
    const int i = id >> 6, q = id & 63;
    const int gph = batch[i];
    const float4 v = ((const float4*)(h + (size_t)i * HID))[q];
    float* p = zin + (size_t)gph * (4 * NHID) + (q << 2);  // cols 0..255
    unsafeAtomicAdd(p + 0, v.x);
    unsafeAtomicAdd(p + 1, v.y);
    unsafeAtomicAdd(p + 2, v.z);
    unsafeAtomicAdd(p + 3, v.w);
    if (q == 0) unsafeAtomicAdd(&cnt[gph], 1.0f);
}

__global__ void pool_finalize(float* __restrict__ zin, const float* __restrict__ cnt) {
    int id = blockIdx.x * blockDim.x + threadIdx.x;        // N_GRAPHS*HID
    if (id >= N_GRAPHS * HID) return;
    const int gph = id >> 8, f = id & 255;
    float* p = zin + (size_t)gph * (4 * NHID) + f;
    *p = selu_f(*p / fmaxf(cnt[gph], 1.0f));
}

__global__ void first_idx_kernel(const int* __restrict__ batch, int* __restrict__ fidx) {
    int gph = threadIdx.x;
    if (gph >= N_GRAPHS) return;
    int lo = 0, hi = N_NODES;                  // lower_bound (searchsorted)
    while (lo < hi) {
        int mid = (lo + hi) >> 1;
        if (batch[mid] < gph) lo = mid + 1; else hi = mid;
    }
    fidx[gph] = lo < N_NODES ? lo : N_NODES - 1;   // jax clamps OOB gathers
}

__global__ void gather_xg(const float* __restrict__ x, const int* __restrict__ fidx,
                          float* __restrict__ xg) {
    int id = blockIdx.x * blockDim.x + threadIdx.x;        // N_GRAPHS*F_IN
    if (id >= N_GRAPHS * F_IN) return;
    const int gph = id >> 8, f = id & 255;
    xg[id] = x[(size_t)fidx[gph] * F_IN + f];
}

__global__ void head_kernel(const float* __restrict__ Z, const float* __restrict__ W,
                            const float* __restrict__ b, float* __restrict__ out) {
    int gph = threadIdx.x;
    if (gph >= N_GRAPHS) return;
    float l0 = b[0], l1 = b[1];
    const float* z = Z + (size_t)gph * NHID;
    #pragma unroll 8
    for (int k = 0; k < NHID; ++k) {
        const float zv = z[k];
        l0 += zv * W[k * 2 + 0];
        l1 += zv * W[k * 2 + 1];
    }
    const float m = fmaxf(l0, l1);
    const float lse = m + logf(expf(l0 - m) + expf(l1 - m));
    out[gph * 2 + 0] = l0 - lse;
    out[gph * 2 + 1] = l1 - lse;
}

// ---------------------------------------------------------------------------
extern "C" void kernel_launch(void* const* d_in, const int* in_sizes, int n_in,
                              void* d_out, int out_size, void* d_ws, size_t ws_size,
                              hipStream_t stream) {
    const float* x    = (const float*)d_in[0];
    const int*   ei   = (const int*)d_in[1];
    const int*   src  = ei;
    const int*   dst  = ei + N_EDGES;
    const int*   batch = (const int*)d_in[2];
    const float* W1   = (const float*)d_in[4];
    const float* b1   = (const float*)d_in[5];
    const float* W2   = (const float*)d_in[6];
    const float* b2   = (const float*)d_in[7];
    const float* fc0w = (const float*)d_in[8];
    const float* fc0b = (const float*)d_in[9];
    const float* fc1w = (const float*)d_in[10];
    const float* fc1b = (const float*)d_in[11];
    const float* fc2w = (const float*)d_in[12];
    const float* fc2b = (const float*)d_in[13];

    char* ws = (char*)d_ws;
    size_t off = 0;
    auto carve = [&](size_t bytes) -> void* {
        void* p = ws + off;
        off = (off + bytes + 255) & ~(size_t)255;
        return p;
    };
    float* bufA = (float*)carve((size_t)N_NODES * HID * 4);   // g = (h@W)*dinv[row]
    float* bufB = (float*)carve((size_t)N_NODES * HID * 4);   // aggregation accumulator
    float* dinv = (float*)carve((size_t)N_NODES * 4);
    float* zin  = (float*)carve((size_t)N_GRAPHS * 4 * NHID * 4);  // [128,512] concat
    float* zfc  = (float*)carve((size_t)N_GRAPHS * NHID * 4);      // [128,128]
    float* cnt  = (float*)carve((size_t)N_GRAPHS * 4);
    int*   fidx = (int*)carve((size_t)N_GRAPHS * 4);
    float* xg   = (float*)carve((size_t)N_GRAPHS * F_IN * 4);

    const int nodeBlocks  = (N_NODES + 255) / 256;
    const int quadBlocksN = (N_NODES * 64 + 255) / 256;
    const int quadBlocksE = (N_EDGES * 64 + 255) / 256;

    // degree + symmetric norm (shared by both convs)
    deg_init <<<nodeBlocks, 256, 0, stream>>>(dinv);
    deg_count<<<(N_EDGES + 255) / 256, 256, 0, stream>>>(dst, dinv);
    deg_rsqrt<<<nodeBlocks, 256, 0, stream>>>(dinv);

    // ---- conv1 : h1 = selu( A_hat @ (x@W1) + b1 ) ----
    wmma_gemm_f32<<<N_NODES / 16, (HID / 64) * 32, 0, stream>>>(
        x, F_IN, W1, HID, bufA, HID, nullptr, dinv, HID, F_IN, /*act=*/0);
    self_init   <<<quadBlocksN, 256, 0, stream>>>(bufA, dinv, bufB);
    edge_scatter<<<quadBlocksE, 256, 0, stream>>>(bufA, dinv, src, dst, bufB);
    bias_selu   <<<quadBlocksN, 256, 0, stream>>>(bufB, b1);      // bufB = h1

    // ---- conv2 : h2 = selu( A_hat @ (h1@W2) + b2 ) ----
    wmma_gemm_f32<<<N_NODES / 16, (HID / 64) * 32, 0, stream>>>(
        bufB, HID, W2, HID, bufA, HID, nullptr, dinv, HID, HID, /*act=*/0);
    self_init   <<<quadBlocksN, 256, 0, stream>>>(bufA, dinv, bufB);
    edge_scatter<<<quadBlocksE, 256, 0, stream>>>(bufA, dinv, src, dst, bufB);
    bias_selu   <<<quadBlocksN, 256, 0, stream>>>(bufB, b2);      // bufB = h2

    // ---- global mean pool -> zin[:, 0:256] ----
    zin_zero     <<<(N_GRAPHS * 4 * NHID + 255) / 256, 256, 0, stream>>>(zin, cnt);
    pool_scatter <<<quadBlocksN, 256, 0, stream>>>(bufB, batch, zin, cnt);
    pool_finalize<<<(N_GRAPHS * HID + 255) / 256, 256, 0, stream>>>(zin, cnt);

    // ---- news = relu(x[first_idx] @ fc0_w + fc0_b) -> zin[:, 256:512] ----
    first_idx_kernel<<<1, 128, 0, stream>>>(batch, fidx);
    gather_xg<<<(N_GRAPHS * F_IN + 255) / 256, 256, 0, stream>>>(x, fidx, xg);
    wmma_gemm_f32<<<N_GRAPHS / 16, (HID / 64) * 32, 0, stream>>>(
        xg, F_IN, fc0w, HID, zin + 256, 4 * NHID, fc0b, nullptr,
        HID, F_IN, /*act=*/1);

    // ---- z = selu(zin @ fc1_w + fc1_b) ----
    wmma_gemm_f32<<<N_GRAPHS / 16, (NHID / 64) * 32, 0, stream>>>(
        zin, 4 * NHID, fc1w, NHID, zfc, NHID, fc1b, nullptr,
        NHID, 4 * NHID, /*act=*/2);

    // ---- logits + log_softmax ----
    head_kernel<<<1, 128, 0, stream>>>(zfc, fc2w, fc2b, (float*)d_out);
}